// FastAttention__86509231276508
// MI455X (gfx1250) — compile-verified
//
#include <hip/hip_runtime.h>
#include <hip/hip_bf16.h>

// Linear attention (chunked causal) for B=4,H=8,N=4096,D=E=64,CHUNK=128.
// 3-kernel plan:
//   k1: per-chunk local KtV (64x64) + colsum(K)  -> ws        (1024 blocks, f32 WMMA)
//   k2: in-place exclusive scan over chunks per (b,h); final Z,S -> d_out tail
//   k3: per-chunk output: Q@S_prev + tril(QKt)@V, scaled by 1/den  (1024 blocks, f32 WMMA)
// Uses V_WMMA_F32_16X16X4_F32 throughout (fp32 precision path).

typedef float v2f __attribute__((ext_vector_type(2)));
typedef float v8f __attribute__((ext_vector_type(8)));

#define NCHUNK   32
#define NBH      32
#define CHUNKSZ  128
#define DDIM     64
#define LSTRIDE  68   // LDS row stride in floats: +4 banks/row, 16B aligned
#define EPSV     1e-6f

__device__ __forceinline__ v8f wmma_f32(v2f a, v2f b, v8f c) {
    return __builtin_amdgcn_wmma_f32_16x16x4_f32(false, a, false, b, (short)0, c, false, false);
}

// ---------------- Kernel 1: per-chunk KtV and colsum(K) ----------------
__global__ __launch_bounds__(256) void k1_chunk_sums(
    const float* __restrict__ K, const float* __restrict__ V,
    float* __restrict__ wsS, float* __restrict__ wsZ)
{
    const int blk = blockIdx.x;
    const int bh = blk / NCHUNK, c = blk % NCHUNK;
    const float* Kc = K + ((size_t)bh * 4096 + (size_t)c * CHUNKSZ) * DDIM;
    const float* Vc = V + ((size_t)bh * 4096 + (size_t)c * CHUNKSZ) * DDIM;

    __shared__ float lK[CHUNKSZ * LSTRIDE];
    __shared__ float lV[CHUNKSZ * LSTRIDE];

    const int tid = threadIdx.x;
    for (int i = tid; i < CHUNKSZ * 16; i += 256) {
        int row = i >> 4, c4 = (i & 15) * 4;
        *(float4*)(lK + row * LSTRIDE + c4) = *(const float4*)(Kc + row * DDIM + c4);
        *(float4*)(lV + row * LSTRIDE + c4) = *(const float4*)(Vc + row * DDIM + c4);
    }
    __syncthreads();

    const int wave = tid >> 5, lane = tid & 31;
    const int m = lane & 15, kh = lane >> 4;

    float* dst = wsS + ((size_t)bh * NCHUNK + c) * 4096;
    // 16 output tiles (4 d-tiles x 4 e-tiles), 2 per wave
    for (int t = 0; t < 2; ++t) {
        const int tile = wave * 2 + t;
        const int d0 = (tile >> 2) * 16, e0 = (tile & 3) * 16;
        v8f acc = {};
        #pragma unroll
        for (int k = 0; k < 32; ++k) {
            const int r = 4 * k + 2 * kh;           // chunk rows (K-dim of GEMM)
            v2f a, b;
            a.x = lK[(r    ) * LSTRIDE + d0 + m];   // A = K^T  (16x4 slice)
            a.y = lK[(r + 1) * LSTRIDE + d0 + m];
            b.x = lV[(r    ) * LSTRIDE + e0 + m];   // B = V    (4x16 slice)
            b.y = lV[(r + 1) * LSTRIDE + e0 + m];
            acc = wmma_f32(a, b, acc);
        }
        #pragma unroll
        for (int r = 0; r < 8; ++r)
            dst[(d0 + r + 8 * kh) * DDIM + e0 + m] = acc[r];
    }
    if (tid < DDIM) {
        float s = 0.f;
        for (int r = 0; r < CHUNKSZ; ++r) s += lK[r * LSTRIDE + tid];
        wsZ[((size_t)bh * NCHUNK + c) * DDIM + tid] = s;
    }
}

// ---------------- Kernel 2: exclusive prefix scan per (b,h) ----------------
__global__ __launch_bounds__(256) void k2_scan(
    float* __restrict__ wsS, float* __restrict__ wsZ,
    float* __restrict__ outZ, float* __restrict__ outS)
{
    const int bh = blockIdx.x, tid = threadIdx.x;
    float acc[16];
    #pragma unroll
    for (int i = 0; i < 16; ++i) acc[i] = 0.f;
    float* base = wsS + (size_t)bh * NCHUNK * 4096;
    for (int c = 0; c < NCHUNK; ++c) {
        float* p = base + (size_t)c * 4096;
        #pragma unroll
        for (int i = 0; i < 16; ++i) {
            const int idx = i * 256 + tid;
            float loc = p[idx];
            p[idx] = acc[i];          // exclusive prefix (S_prev for chunk c)
            acc[i] += loc;
        }
    }
    #pragma unroll
    for (int i = 0; i < 16; ++i) outS[(size_t)bh * 4096 + i * 256 + tid] = acc[i];

    if (tid < DDIM) {
        float az = 0.f;
        float* zb = wsZ + (size_t)bh * NCHUNK * DDIM;
        for (int c = 0; c < NCHUNK; ++c) {
            float loc = zb[c * DDIM + tid];
            zb[c * DDIM + tid] = az;
            az += loc;
        }
        outZ[bh * DDIM + tid] = az;
    }
}

// ---------------- Kernel 3: per-chunk outputs ----------------
__global__ __launch_bounds__(256) void k3_output(
    const float* __restrict__ Q, const float* __restrict__ K, const float* __restrict__ V,
    const float* __restrict__ wsS, const float* __restrict__ wsZ, float* __restrict__ out)
{
    const int blk = blockIdx.x;
    const int bh = blk / NCHUNK, c = blk % NCHUNK;
    const size_t cbase = ((size_t)bh * 4096 + (size_t)c * CHUNKSZ) * DDIM;
    const float* Qc = Q + cbase;
    const float* Kc = K + cbase;
    const float* Vc = V + cbase;
    const float* Sp = wsS + ((size_t)bh * NCHUNK + c) * 4096;
    const float* Zp = wsZ + ((size_t)bh * NCHUNK + c) * DDIM;

    __shared__ float lQ[CHUNKSZ * LSTRIDE];
    __shared__ float lK[CHUNKSZ * LSTRIDE];
    __shared__ float lV[CHUNKSZ * LSTRIDE];
    __shared__ float lS[DDIM * LSTRIDE];
    __shared__ float lZ[DDIM];
    __shared__ float lA[8][16 * 20];   // per-wave masked-A staging (C-layout -> A-operand)
    __shared__ float lDen[CHUNKSZ];

    const int tid = threadIdx.x;
    for (int i = tid; i < CHUNKSZ * 16; i += 256) {
        int row = i >> 4, c4 = (i & 15) * 4;
        *(float4*)(lQ + row * LSTRIDE + c4) = *(const float4*)(Qc + row * DDIM + c4);
        *(float4*)(lK + row * LSTRIDE + c4) = *(const float4*)(Kc + row * DDIM + c4);
        *(float4*)(lV + row * LSTRIDE + c4) = *(const float4*)(Vc + row * DDIM + c4);
    }
    for (int i = tid; i < DDIM * 16; i += 256) {
        int row = i >> 4, c4 = (i & 15) * 4;
        *(float4*)(lS + row * LSTRIDE + c4) = *(const float4*)(Sp + row * DDIM + c4);
    }
    if (tid < DDIM) lZ[tid] = Zp[tid];
    __syncthreads();

    const int wave = tid >> 5, lane = tid & 31;
    const int m = lane & 15, kh = lane >> 4;
    const int r0 = wave * 16;                  // this wave's 16-row strip

    v8f acc[4] = { v8f{}, v8f{}, v8f{}, v8f{} };  // out e-tiles (16x16 each)
    v8f rsum = {};                                // elementwise sum of masked A tiles

    // ---- inter: Q_strip (16x64) @ S_prev (64x64) ----
    #pragma unroll
    for (int et = 0; et < 4; ++et) {
        #pragma unroll
        for (int k = 0; k < 16; ++k) {
            const int d = 4 * k + 2 * kh;
            v2f a = *(const v2f*)(lQ + (r0 + m) * LSTRIDE + d);   // contiguous pair
            v2f b;
            b.x = lS[(d    ) * LSTRIDE + et * 16 + m];
            b.y = lS[(d + 1) * LSTRIDE + et * 16 + m];
            acc[et] = wmma_f32(a, b, acc[et]);
        }
    }

    // ---- intra: A = Q K^T (causal), out += tril(A) @ V ----
    for (int j = 0; j <= wave; ++j) {          // column tiles above diagonal are all-zero
        v8f a8 = {};
        #pragma unroll
        for (int k = 0; k < 16; ++k) {
            const int d = 4 * k + 2 * kh;
            v2f a = *(const v2f*)(lQ + (r0 + m)    * LSTRIDE + d);
            v2f b = *(const v2f*)(lK + (j * 16 + m) * LSTRIDE + d);  // B = K^T slice
            a8 = wmma_f32(a, b, a8);
        }
        if (j == wave) {                        // diagonal tile: keep col <= row
            #pragma unroll
            for (int r = 0; r < 8; ++r) {
                const int row = r + 8 * kh;
                if (m > row) a8[r] = 0.f;
            }
        }
        rsum += a8;

        // stage masked A tile (C/D layout) into per-wave LDS buffer as row-major 16x16
        float* Ab = &lA[wave][0];
        #pragma unroll
        for (int r = 0; r < 8; ++r) Ab[(r + 8 * kh) * 20 + m] = a8[r];
        // same-wave LDS is in-order: no barrier needed before reload

        #pragma unroll
        for (int et = 0; et < 4; ++et) {
            #pragma unroll
            for (int ks = 0; ks < 4; ++ks) {
                const int kk = 4 * ks + 2 * kh;
                v2f a = *(const v2f*)(Ab + m * 20 + kk);
                v2f b;
                b.x = lV[(j * 16 + kk    ) * LSTRIDE + et * 16 + m];
                b.y = lV[(j * 16 + kk + 1) * LSTRIDE + et * 16 + m];
                acc[et] = wmma_f32(a, b, acc[et]);
            }
        }
    }

    // ---- denominators: rowsum(tril A) + sum_d q*(Z_prev + eps) ----
    float rs[8];
    #pragma unroll
    for (int r = 0; r < 8; ++r) {
        float vv = rsum[r];
        vv += __shfl_xor(vv, 1, 32);
        vv += __shfl_xor(vv, 2, 32);
        vv += __shfl_xor(vv, 4, 32);
        vv += __shfl_xor(vv, 8, 32);           // reduce within each 16-lane half
        rs[r] = vv;                             // row = r + 8*kh
    }
    if (m == 0) {
        #pragma unroll
        for (int r = 0; r < 8; ++r) lDen[r0 + r + 8 * kh] = rs[r];
    }
    if (kh == 0) {                              // lanes 0..15: one row each
        float qz = 0.f;
        for (int d = 0; d < DDIM; ++d)
            qz += lQ[(r0 + m) * LSTRIDE + d] * (lZ[d] + EPSV);
        lDen[r0 + m] = 1.0f / (qz + lDen[r0 + m]);
    }

    // ---- scale + store ----
    float* oc = out + cbase;
    #pragma unroll
    for (int et = 0; et < 4; ++et) {
        #pragma unroll
        for (int r = 0; r < 8; ++r) {
            const int row = r + 8 * kh;
            oc[(r0 + row) * DDIM + et * 16 + m] = acc[et][r] * lDen[r0 + row];
        }
    }
}

extern "C" void kernel_launch(void* const* d_in, const int* in_sizes, int n_in,
                              void* d_out, int out_size, void* d_ws, size_t ws_size,
                              hipStream_t stream) {
    const float* q = (const float*)d_in[0];
    const float* k = (const float*)d_in[1];
    const float* v = (const float*)d_in[2];
    // d_in[3] = inferrring_states (unused scalar)

    float* out = (float*)d_out;
    float* outZ = out + (size_t)NBH * 4096 * DDIM;   // 8,388,608 floats of `out`
    float* outS = outZ + NBH * DDIM;                 // then 2048 of Z, then 131072 of S

    // workspace: per-(bh,chunk) S (64x64) and Z (64); scanned in place. ~16.25 MB.
    float* wsS = (float*)d_ws;
    float* wsZ = wsS + (size_t)NBH * NCHUNK * 4096;

    k1_chunk_sums<<<NBH * NCHUNK, 256, 0, stream>>>(k, v, wsS, wsZ);
    k2_scan<<<NBH, 256, 0, stream>>>(wsS, wsZ, outZ, outS);
    k3_output<<<NBH * NCHUNK, 256, 0, stream>>>(q, k, v, wsS, wsZ, out);
}